// KappaGCN_29240137351292
// MI455X (gfx1250) — compile-verified
//
#include <hip/hip_runtime.h>

typedef __attribute__((ext_vector_type(2))) float v2f;
typedef __attribute__((ext_vector_type(8))) float v8f;
typedef __attribute__((address_space(3))) char lds_char_t;

#define DIM 32
#define NC 16
#define NCOLP 48   // padded fused-B columns: 32 (gamma*XW) + 1 (gamma-1) + 1 (ones) + 14 pad

__device__ __forceinline__ float artanh_clip(float x) {
    x = fminf(fmaxf(x, -1.0f + 1e-7f), 1.0f - 1e-7f);
    return 0.5f * logf((1.0f + x) / (1.0f - x));
}

// async 16B global -> LDS copy (per lane), tracked by ASYNCcnt
__device__ __forceinline__ void async_b128(float* lds, const float* gaddr) {
    unsigned lo = (unsigned)(size_t)(lds_char_t*)lds;   // LDS byte offset
    asm volatile("global_load_async_to_lds_b128 %0, %1, off"
                 :: "v"(lo), "v"(gaddr) : "memory");
}

template <int Np>
__device__ __forceinline__ void wait_async() {
#if __has_builtin(__builtin_amdgcn_s_wait_asynccnt)
    __builtin_amdgcn_s_wait_asynccnt(Np);
#else
    asm volatile("s_wait_asynccnt %0" :: "n"(Np) : "memory");
#endif
}

// ---------------------------------------------------------------------------
// prep: XW = mobius_matvec(W, X); gamma = lambda(XW);
// Bt (transposed, [48][N]): rows 0..31 = gamma*XW, 32 = gamma-1, 33 = 1, rest 0
// ---------------------------------------------------------------------------
__global__ void kappa_prep(const float* __restrict__ X, const float* __restrict__ W,
                           float* __restrict__ Bt, int N) {
    __shared__ float sW[DIM * DIM];
    for (int i = threadIdx.x; i < DIM * DIM; i += blockDim.x) sW[i] = W[i];
    __syncthreads();
    int n = blockIdx.x * blockDim.x + threadIdx.x;
    if (n >= N) return;

    float x[DIM]; float xs = 0.f;
#pragma unroll
    for (int i = 0; i < DIM; ++i) { x[i] = X[n * DIM + i]; xs += x[i] * x[i]; }
    float xn = sqrtf(fmaxf(xs, 1e-30f));

    float mx[DIM]; float ms = 0.f;
#pragma unroll
    for (int j = 0; j < DIM; ++j) {
        float acc = 0.f;
        for (int i = 0; i < DIM; ++i) acc += x[i] * sW[i * DIM + j];
        mx[j] = acc; ms += acc * acc;
    }
    float mxn = sqrtf(fmaxf(ms, 1e-30f));
    float t = tanhf(mxn / xn * artanh_clip(xn)) / mxn;   // mobius_matvec scale

    float g2 = 0.f;
#pragma unroll
    for (int j = 0; j < DIM; ++j) { mx[j] *= t; g2 += mx[j] * mx[j]; }
    float gamma = 2.0f / fmaxf(1.0f - g2, 1e-15f);       // lambda_x, K=-1

#pragma unroll
    for (int j = 0; j < DIM; ++j) Bt[(size_t)j * N + n] = gamma * mx[j];
    Bt[(size_t)DIM * N + n]       = gamma - 1.0f;
    Bt[(size_t)(DIM + 1) * N + n] = 1.0f;
#pragma unroll
    for (int j = DIM + 2; j < NCOLP; ++j) Bt[(size_t)j * N + n] = 0.f;
}

// ---------------------------------------------------------------------------
// post: Y[n] = {nom[0..31], denom, alpha}; gyromidpoint + lincomb + sigma
// ---------------------------------------------------------------------------
__global__ void kappa_post(const float* __restrict__ Y, float* __restrict__ Xout, int N) {
    int n = blockIdx.x * blockDim.x + threadIdx.x;
    if (n >= N) return;
    const float* y = Y + (size_t)n * NCOLP;

    float denom = y[DIM];
    denom = (denom >= 0.f) ? fmaxf(denom, 1e-10f) : fminf(denom, -1e-10f);
    float alpha = y[DIM + 1];

    float v[DIM]; float vs = 0.f;
#pragma unroll
    for (int d = 0; d < DIM; ++d) { v[d] = y[d] / denom; vs += v[d] * v[d]; }
    float vn = sqrtf(fmaxf(vs, 1e-30f));
    float s1 = tanhf(0.5f * artanh_clip(vn)) / vn;        // mobius_scalar_mul(0.5)

    float as = s1 * s1 * vs;
    float an = sqrtf(fmaxf(as, 1e-30f));
    float s2 = tanhf(alpha * artanh_clip(an)) / an;       // mobius_scalar_mul(alpha)

    float s12 = s1 * s2;
    float wsum = s12 * s12 * vs;
    float wn = sqrtf(fmaxf(wsum, 1e-30f));
    float s3 = artanh_clip(wn) / wn;                      // logmap0 scale
    float s123 = s3 * s12;

    float u[DIM]; float us = 0.f;
#pragma unroll
    for (int d = 0; d < DIM; ++d) { u[d] = fmaxf(s123 * v[d], 0.f); us += u[d] * u[d]; } // relu
    float un = sqrtf(fmaxf(us, 1e-30f));
    float s4 = tanhf(un) / un;                            // expmap0
#pragma unroll
    for (int d = 0; d < DIM; ++d) Xout[(size_t)n * DIM + d] = s4 * u[d];
}

// ---------------------------------------------------------------------------
// logits: hyperbolic MLR; writes Lt transposed [16][N] for the final GEMM
// ---------------------------------------------------------------------------
__global__ void kappa_logits(const float* __restrict__ X, const float* __restrict__ Wl,
                             const float* __restrict__ pk, float* __restrict__ Lt, int N) {
    __shared__ float sW[DIM * NC];   // Wl row-major [d][c]
    __shared__ float sB[NC * DIM];   // p_ks [c][d]
    __shared__ float sAn[NC], sLam[NC];
    for (int i = threadIdx.x; i < DIM * NC; i += blockDim.x) { sW[i] = Wl[i]; sB[i] = pk[i]; }
    __syncthreads();
    if (threadIdx.x < NC) {
        int c = threadIdx.x;
        float a2 = 0.f, b2 = 0.f;
        for (int d = 0; d < DIM; ++d) {
            float w = sW[d * NC + c]; a2 += w * w;
            float b = sB[c * DIM + d]; b2 += b * b;
        }
        sAn[c] = fmaxf(sqrtf(a2), 1e-10f);
        sLam[c] = 2.0f / fmaxf(1.0f - b2, 1e-15f);
    }
    __syncthreads();

    int n = blockIdx.x * blockDim.x + threadIdx.x;
    if (n >= N) return;
    float x[DIM]; float y2 = 0.f;
#pragma unroll
    for (int d = 0; d < DIM; ++d) { x[d] = X[(size_t)n * DIM + d]; y2 += x[d] * x[d]; }

    for (int c = 0; c < NC; ++c) {
        float b2 = 0.f, xb = 0.f;
        for (int d = 0; d < DIM; ++d) {
            float b = sB[c * DIM + d];
            b2 += b * b; xb += b * x[d];
        }
        // mobius_add(-b, x), K=-1
        float coefA = 1.0f - 2.0f * xb + y2;
        float coefB = 1.0f - b2;
        float den = fmaxf(1.0f - 2.0f * xb + b2 * y2, 1e-15f);
        float inv = 1.0f / den;
        float zn2 = 0.f, za = 0.f;
        for (int d = 0; d < DIM; ++d) {
            float z = (coefA * (-sB[c * DIM + d]) + coefB * x[d]) * inv;
            zn2 += z * z;
            za += z * sW[d * NC + c];
        }
        float zn = fmaxf(sqrtf(fmaxf(zn2, 1e-30f)), 1e-10f);
        float dist = asinhf(2.0f * za / ((1.0f - zn * zn) * sAn[c]));
        Lt[(size_t)c * N + n] = sLam[c] * sAn[c] * dist;
    }
}

// ---------------------------------------------------------------------------
// Y[N x 16*NTC] = A[N x N] @ Bt^T, Bt stored [16*NTC][N] (row = output column)
// fp32 WMMA 16x16x4; K staged via async-to-LDS double buffering (ASYNCcnt)
// ---------------------------------------------------------------------------
template <int NTC, int NTR>
__global__ void __launch_bounds__(32 * NTR * NTC)
agemm(const float* __restrict__ A, const float* __restrict__ Bt,
      float* __restrict__ Y, int N) {
    constexpr int ROWS = NTR * 16;
    constexpr int KC = 32;
    constexpr int Q = KC / 4;             // b128 units per row
    constexpr int PAD = KC + 4;           // 36: rows = 144B (16B aligned), bank-conflict-free
    constexpr int NCOL = NTC * 16;
    constexpr int NW = NTR * NTC;         // waves per block
    constexpr int AU = ROWS * Q;          // A copy units (16B each)
    constexpr int U = (ROWS + NCOL) * Q;  // total copy units (multiple of 32)
    constexpr int GROUPS = U / 32;        // full-EXEC 32-lane copy groups
    constexpr int MINTRIPS = GROUPS / NW; // min async issues per wave per stage

    __shared__ float sA[2][ROWS * PAD];
    __shared__ float sB[2][NCOL * PAD];

    const int tid = threadIdx.x;
    const int wave = tid >> 5;
    const int lane = tid & 31;
    const int tr = wave / NTC;
    const int tc = wave % NTC;
    const int half = lane >> 4;
    const int l16 = lane & 15;
    const int rowBase = blockIdx.x * ROWS;

    const int aoff = (tr * 16 + l16) * PAD + half * 2;
    const int boff = (tc * 16 + l16) * PAD + half * 2;

    // stage one K-chunk into buffer `buf` via async LDS copies (whole-wave groups)
    auto stage = [&](int buf, int k0) {
        for (int g = wave; g < GROUPS; g += NW) {   // wave-uniform trip count
            int u = g * 32 + lane;
            const float* gp; float* lp;
            if (u < AU) {
                int r = u / Q, c4 = (u % Q) * 4;
                gp = A + (size_t)(rowBase + r) * N + (k0 + c4);
                lp = &sA[buf][r * PAD + c4];
            } else {
                int v = u - AU;
                int r = v / Q, c4 = (v % Q) * 4;
                gp = Bt + (size_t)r * N + (k0 + c4);
                lp = &sB[buf][r * PAD + c4];
            }
            async_b128(lp, gp);
        }
    };

    v8f acc = {};
    stage(0, 0);
    int p = 0;
    for (int k0 = 0; k0 < N; k0 += KC) {
        __syncthreads();                      // everyone done reading buf 1-p
        if (k0 + KC < N) {
            stage(1 - p, k0 + KC);            // prefetch next chunk (stays in flight)
            wait_async<MINTRIPS>();           // drains chunk p (in-order completion)
        } else {
            wait_async<0>();
        }
        __syncthreads();                      // buf p visible to all waves

#pragma unroll
        for (int kk = 0; kk < KC; kk += 4) {
            v2f af = *(const v2f*)&sA[p][aoff + kk];
            v2f bf = *(const v2f*)&sB[p][boff + kk];
            acc = __builtin_amdgcn_wmma_f32_16x16x4_f32(
                false, af, false, bf, (short)0, acc, false, false);
        }
        p ^= 1;
    }

    // C/D layout: VGPR i -> lanes0-15: M=i, lanes16-31: M=8+i; N=l16
    const int m0 = rowBase + tr * 16 + half * 8;
    const int nc = tc * 16 + l16;
#pragma unroll
    for (int i = 0; i < 8; ++i)
        Y[(size_t)(m0 + i) * NCOL + nc] = acc[i];
}

// ---------------------------------------------------------------------------
extern "C" void kernel_launch(void* const* d_in, const int* in_sizes, int n_in,
                              void* d_out, int out_size, void* d_ws, size_t ws_size,
                              hipStream_t stream) {
    const float* X  = (const float*)d_in[0];
    const float* A  = (const float*)d_in[1];
    const float* W1 = (const float*)d_in[2];
    const float* W2 = (const float*)d_in[3];
    const float* Wl = (const float*)d_in[4];
    const float* pk = (const float*)d_in[5];
    float* out = (float*)d_out;
    const int N = in_sizes[0] / DIM;   // 8192

    float* ws = (float*)d_ws;
    float* Bt = ws;                          // [48][N]
    float* Yb = Bt + (size_t)NCOLP * N;      // [N][48]
    float* Xa = Yb + (size_t)N * NCOLP;      // [N][32]
    float* Xb = Xa + (size_t)N * DIM;        // [N][32]
    float* Lt = Xb + (size_t)N * DIM;        // [16][N]

    const int ew = 256;
    const int eb = (N + ew - 1) / ew;
    const int gb = N / 32;                   // 256 row blocks (32 rows each)

    // layer 1
    kappa_prep  <<<eb, ew, 0, stream>>>(X, W1, Bt, N);
    agemm<3, 2> <<<gb, 192, 0, stream>>>(A, Bt, Yb, N);
    kappa_post  <<<eb, ew, 0, stream>>>(Yb, Xa, N);
    // layer 2
    kappa_prep  <<<eb, ew, 0, stream>>>(Xa, W2, Bt, N);
    agemm<3, 2> <<<gb, 192, 0, stream>>>(A, Bt, Yb, N);
    kappa_post  <<<eb, ew, 0, stream>>>(Yb, Xb, N);
    // logits + aggregation (writes d_out directly, stride 16)
    kappa_logits<<<eb, ew, 0, stream>>>(Xb, Wl, pk, Lt, N);
    agemm<1, 2> <<<gb, 64, 0, stream>>>(A, Lt, out, N);
}